// MoleculeEmbedding_9174050144966
// MI455X (gfx1250) — compile-verified
//
#include <hip/hip_runtime.h>
#include <math.h>

// Problem constants (from the reference)
#define Gn 4096
#define Nn 131072
#define Fn 256
#define F3 768

typedef __attribute__((ext_vector_type(2))) float v2f;
typedef __attribute__((ext_vector_type(8))) float v8f;

__device__ __forceinline__ v2f ldg2(const float* p) {
    return *reinterpret_cast<const v2f*>(p);
}

// ---- mark which graphs have at least one node (only place batch/x matter) ----
__global__ void zero_counts_kernel(int* counts) {
    int i = blockIdx.x * blockDim.x + threadIdx.x;
    if (i < Gn) counts[i] = 0;
}
__global__ void mark_counts_kernel(const int* __restrict__ batch, int* counts) {
    int i = blockIdx.x * blockDim.x + threadIdx.x;
    if (i < Nn) counts[batch[i]] = 1;   // idempotent store, race-free
}

// ---- one message-passing layer, fully fused ----
// Exploits: attended rows are identical within a graph and softmax weights sum
// to 1, so c = elu(h_in @ atw^T + atb) (0 for empty graphs); then
// h_out = GRUCell(c, h_in) with torch gate order [r,z,n].
__launch_bounds__(512)
__global__ void layer_kernel(const float* __restrict__ h_in,   // [G,F]
                             const float* __restrict__ atw,    // [F,F] row-major [out,in]
                             const float* __restrict__ atb,    // [F]
                             const float* __restrict__ wih,    // [3F,F]
                             const float* __restrict__ bih,    // [3F]
                             const float* __restrict__ whh,    // [3F,F]
                             const float* __restrict__ bhh,    // [3F]
                             const int*   __restrict__ nonempty, // [G]
                             float* __restrict__ h_out)        // [G,F]
{
    __shared__ float c_lds[16 * Fn];            // 16 rows x 256 cols of c (16 KB)

    const int lane  = threadIdx.x & 31;
    const int wv    = threadIdx.x >> 5;         // wave id 0..15 -> column tile
    const int r0    = blockIdx.x * 16;          // graph row base for this block
    const int col0  = wv * 16;
    const int frow  = lane & 15;                // M for A-fragments, N for B-fragments
    const int khalf = (lane >> 4) * 2;          // lanes 0-15: K {0,1}; lanes 16-31: K {2,3}

    // -------- Phase 1: c tile = elu(h_in @ atw^T + atb), masked --------
    v8f acc = {};
    const float* aRow = h_in + (size_t)(r0 + frow) * Fn + khalf;
    const float* bRow = atw  + (size_t)(col0 + frow) * Fn + khalf;
#pragma unroll 4
    for (int k0 = 0; k0 < Fn; k0 += 4) {
        v2f a = ldg2(aRow + k0);
        v2f b = ldg2(bRow + k0);
        acc = __builtin_amdgcn_wmma_f32_16x16x4_f32(false, a, false, b,
                                                    (short)0, acc, false, false);
    }
    {
        const int   ncol  = col0 + frow;        // D layout: N = lane % 16
        const float bias  = atb[ncol];
        const int   mbase = (lane >> 4) * 8;    // D layout: M = vgpr + 8*(lane>=16)
#pragma unroll
        for (int v = 0; v < 8; ++v) {
            int   M   = mbase + v;
            float val = acc[v] + bias;
            val = val > 0.0f ? val : expm1f(val);                  // ELU (matches jax expm1)
            val = nonempty[r0 + M] ? val : 0.0f;                    // empty graph -> c = 0
            c_lds[M * Fn + ncol] = val;
        }
    }
    __syncthreads();

    // -------- Phase 2: GRU gates, 6 WMMA accumulation chains --------
    v8f air = {}, aiz = {}, ain = {}, ahr = {}, ahz = {}, ahn = {};
    const float* cRow = &c_lds[frow * Fn + khalf];
    const float* hRow = h_in + (size_t)(r0 + frow) * Fn + khalf;
    const float* wir  = wih + (size_t)(       col0 + frow) * Fn + khalf;
    const float* wiz  = wih + (size_t)(Fn   + col0 + frow) * Fn + khalf;
    const float* win  = wih + (size_t)(2*Fn + col0 + frow) * Fn + khalf;
    const float* whr  = whh + (size_t)(       col0 + frow) * Fn + khalf;
    const float* whz  = whh + (size_t)(Fn   + col0 + frow) * Fn + khalf;
    const float* whn  = whh + (size_t)(2*Fn + col0 + frow) * Fn + khalf;
#pragma unroll 2
    for (int k0 = 0; k0 < Fn; k0 += 4) {
        v2f ac = *reinterpret_cast<const v2f*>(cRow + k0);   // LDS read (ds_load_b64)
        v2f ah = ldg2(hRow + k0);
        v2f b0 = ldg2(wir + k0);
        v2f b1 = ldg2(wiz + k0);
        v2f b2 = ldg2(win + k0);
        v2f b3 = ldg2(whr + k0);
        v2f b4 = ldg2(whz + k0);
        v2f b5 = ldg2(whn + k0);
        air = __builtin_amdgcn_wmma_f32_16x16x4_f32(false, ac, false, b0, (short)0, air, false, false);
        aiz = __builtin_amdgcn_wmma_f32_16x16x4_f32(false, ac, false, b1, (short)0, aiz, false, false);
        ain = __builtin_amdgcn_wmma_f32_16x16x4_f32(false, ac, false, b2, (short)0, ain, false, false);
        ahr = __builtin_amdgcn_wmma_f32_16x16x4_f32(false, ah, false, b3, (short)0, ahr, false, false);
        ahz = __builtin_amdgcn_wmma_f32_16x16x4_f32(false, ah, false, b4, (short)0, ahz, false, false);
        ahn = __builtin_amdgcn_wmma_f32_16x16x4_f32(false, ah, false, b5, (short)0, ahn, false, false);
    }
    {
        const int   ncol = col0 + frow;
        const float bir = bih[ncol], biz = bih[Fn + ncol], bin = bih[2*Fn + ncol];
        const float bhr = bhh[ncol], bhz = bhh[Fn + ncol], bhn = bhh[2*Fn + ncol];
        const int   mbase = (lane >> 4) * 8;
#pragma unroll
        for (int v = 0; v < 8; ++v) {
            int   g    = r0 + mbase + v;
            float hold = h_in[(size_t)g * Fn + ncol];
            float r  = 1.0f / (1.0f + expf(-(air[v] + bir + ahr[v] + bhr)));
            float z  = 1.0f / (1.0f + expf(-(aiz[v] + biz + ahz[v] + bhz)));
            float nn = tanhf(ain[v] + bin + r * (ahn[v] + bhn));
            h_out[(size_t)g * Fn + ncol] = (1.0f - z) * nn + z * hold;
        }
    }
}

extern "C" void kernel_launch(void* const* d_in, const int* in_sizes, int n_in,
                              void* d_out, int out_size, void* d_ws, size_t ws_size,
                              hipStream_t stream) {
    (void)in_sizes; (void)n_in; (void)out_size; (void)ws_size;

    const float* h_s      = (const float*)d_in[0];
    // d_in[1] = x        : unused (cancels out: softmax weights sum to 1)
    const int*   batch    = (const int*)d_in[2];
    // d_in[3], d_in[4]   : align_w / align_b, unused for the same reason
    const float* attend_w = (const float*)d_in[5];  // [T,F,F]
    const float* attend_b = (const float*)d_in[6];  // [T,F]
    const float* w_ih     = (const float*)d_in[7];  // [T,3F,F]
    const float* b_ih     = (const float*)d_in[8];  // [T,3F]
    const float* w_hh     = (const float*)d_in[9];  // [T,3F,F]
    const float* b_hh     = (const float*)d_in[10]; // [T,3F]

    int*   counts = (int*)d_ws;                       // Gn ints
    float* h1     = (float*)((char*)d_ws + 16384);    // [G,F] intermediate h

    zero_counts_kernel<<<Gn / 256, 256, 0, stream>>>(counts);
    mark_counts_kernel<<<Nn / 256, 256, 0, stream>>>(batch, counts);

    // layer 0: h_s -> h1
    layer_kernel<<<Gn / 16, 512, 0, stream>>>(h_s,
                                              attend_w, attend_b,
                                              w_ih, b_ih, w_hh, b_hh,
                                              counts, h1);
    // layer 1: h1 -> d_out
    layer_kernel<<<Gn / 16, 512, 0, stream>>>(h1,
                                              attend_w + Fn * Fn, attend_b + Fn,
                                              w_ih + F3 * Fn, b_ih + F3,
                                              w_hh + F3 * Fn, b_hh + F3,
                                              counts, (float*)d_out);
}